// FastGraphConvolution_56556129354713
// MI455X (gfx1250) — compile-verified
//
#include <hip/hip_runtime.h>
#include <stdint.h>

// ---------------------------------------------------------------------------
// FastGraphConvolution for gfx1250 (MI455X)
//   N=100000 nodes, D=32 neighbors, F_IN=F_OUT=128, S=16 samples
//   Stage A: importance (degree) reduction
//   Stage B: per-wave sampling + gather-aggregate (L2-resident H)
//   Stage C: [16x128] x [128x128] GEMM via V_WMMA_F32_16X16X4_F32 (fp32)
//   W is staged TRANSPOSED in LDS so each B-fragment is one ds_load_b64
//   into an even-aligned VGPR pair (no repack movs, conflict-free banks).
// ---------------------------------------------------------------------------

typedef __attribute__((ext_vector_type(2))) float v2f;
typedef __attribute__((ext_vector_type(8))) float v8f;

#define DEG   32
#define FIN   128
#define FOUT  128
#define SS    16
#define EPSF  1e-16f

// LDS layout (floats):
//   sWt  : FOUT cols x TSTR   (transposed W: sWt[n][k] = W[k][n]; stride 132)
//   sB   : FOUT bias
//   sAgg : per-wave 16 x ASTR (stride 132 -> conflict-free A-frag b64 reads)
constexpr int TSTR = 132;
constexpr int ASTR = 132;
constexpr int WAVES_PER_BLOCK = 8;
constexpr int SWT_FLOATS = FOUT * TSTR;                // 16896
constexpr int SB_FLOATS  = FOUT;                       // 128
constexpr int AGG_FLOATS = 16 * ASTR;                  // 2112 per wave
constexpr int SMEM_FLOATS = SWT_FLOATS + SB_FLOATS + WAVES_PER_BLOCK * AGG_FLOATS;
constexpr size_t SMEM_BYTES = (size_t)SMEM_FLOATS * sizeof(float); // ~133 KB

__device__ __forceinline__ unsigned pcg_hash(unsigned v) {
    v = v * 747796405u + 2891336453u;
    unsigned w = ((v >> ((v >> 28u) + 4u)) ^ v) * 277803737u;
    return (w >> 22u) ^ w;
}

// ---------------------------------------------------------------------------
// Kernel 1: degrees[i] = sum_j edge_vals[i][j] + EPS; block partial sums
// ---------------------------------------------------------------------------
__global__ void fgc_degree_kernel(const float* __restrict__ edge_vals,
                                  float* __restrict__ imp,
                                  float* __restrict__ partials, int n) {
    __shared__ float red[256];
    const int i = blockIdx.x * 256 + threadIdx.x;
    float s = 0.0f;
    if (i < n) {
        const float4* row = (const float4*)(edge_vals + (size_t)i * DEG);
        float acc = 0.0f;
#pragma unroll
        for (int j = 0; j < DEG / 4; ++j) {
            float4 v = row[j];
            acc += v.x + v.y + v.z + v.w;
        }
        s = acc + EPSF;
        imp[i] = s;
    }
    red[threadIdx.x] = s;
    __syncthreads();
#pragma unroll
    for (int off = 128; off > 0; off >>= 1) {
        if (threadIdx.x < off) red[threadIdx.x] += red[threadIdx.x + off];
        __syncthreads();
    }
    if (threadIdx.x == 0) partials[blockIdx.x] = red[0];
}

// ---------------------------------------------------------------------------
// Kernel 2: single-block deterministic reduction of partials -> total
// ---------------------------------------------------------------------------
__global__ void fgc_reduce_kernel(const float* __restrict__ partials, int np,
                                  float* __restrict__ total) {
    __shared__ float red[256];
    float s = 0.0f;
    for (int i = threadIdx.x; i < np; i += 256) s += partials[i];
    red[threadIdx.x] = s;
    __syncthreads();
#pragma unroll
    for (int off = 128; off > 0; off >>= 1) {
        if (threadIdx.x < off) red[threadIdx.x] += red[threadIdx.x + off];
        __syncthreads();
    }
    if (threadIdx.x == 0) total[0] = red[0];
}

// ---------------------------------------------------------------------------
// Kernel 3: per-wave sample+aggregate 16 nodes, then WMMA GEMM with W
// ---------------------------------------------------------------------------
__global__ __launch_bounds__(WAVES_PER_BLOCK * 32, 1)
void fgc_main_kernel(const float* __restrict__ H,
                     const int* __restrict__ neighbors,
                     const float* __restrict__ W,
                     const float* __restrict__ b,
                     const float* __restrict__ imp,
                     const float* __restrict__ totalp,
                     float* __restrict__ out, int n, int ntiles) {
    extern __shared__ float smem[];
    float* sWt  = smem;                     // transposed W
    float* sB   = smem + SWT_FLOATS;
    float* sAgg = sB + SB_FLOATS;

    const int tid  = threadIdx.x;
    const int lane = tid & 31;
    const int wave = tid >> 5;

    // ---- Stage W (transposed) + bias into LDS ----
    // 256 threads: thread t handles W row (t>>1), column half (t&1)*64..+63.
    // Write W[k][c] -> sWt[c*TSTR + k]; per-wave column writes hit 16
    // consecutive banks twice (c, c+64 alias: 64*TSTR ≡ 0 mod 64) -> 2-way conflict.
    {
        const int krow = tid >> 1;
        const int cbase = (tid & 1) * 64;
        const float4* src = (const float4*)(W + (size_t)krow * FOUT + cbase);
#pragma unroll
        for (int c4 = 0; c4 < 16; ++c4) {
            float4 v = src[c4];
            const int c = cbase + 4 * c4;
            sWt[(c + 0) * TSTR + krow] = v.x;
            sWt[(c + 1) * TSTR + krow] = v.y;
            sWt[(c + 2) * TSTR + krow] = v.z;
            sWt[(c + 3) * TSTR + krow] = v.w;
        }
        if (tid < FOUT) sB[tid] = b[tid];
    }
    __syncthreads();

    const float invTotal = 1.0f / totalp[0];
    const int tile = blockIdx.x * WAVES_PER_BLOCK + wave;
    if (tile >= ntiles) return;            // wave-uniform exit, after barrier

    float* agg = sAgg + wave * AGG_FLOATS;

    // ---- Stage 1: sampling + weighted aggregation (16 nodes per wave) ----
    for (int m = 0; m < 16; ++m) {
        const int g  = tile * 16 + m;
        const int gc = (g < n) ? g : (n - 1);

        const int   nbr = neighbors[(size_t)gc * DEG + lane]; // lane <-> neighbor slot
        const float p   = imp[nbr] * invTotal;                 // global prob

        // inclusive prefix scan over 32 lanes -> CDF
        float cdf = p;
#pragma unroll
        for (int off = 1; off < 32; off <<= 1) {
            float o = __shfl_up(cdf, off, 32);
            if (lane >= off) cdf += o;
        }
        const float rowsum = __shfl(cdf, 31, 32);
        const float invrs  = 1.0f / (rowsum + EPSF);
        const float nprob  = p * invrs;     // row-normalized prob of this lane's slot
        const float cdfn   = cdf * invrs;   // normalized CDF (~1 at lane 31)

        float4 acc = make_float4(0.0f, 0.0f, 0.0f, 0.0f);
#pragma unroll
        for (int s = 0; s < SS; ++s) {
            // deterministic per-(node, draw) uniform in [0,1)
            const unsigned hh = pcg_hash((unsigned)g * 197u + (unsigned)s * 16807u + 0x9E3779B9u);
            const float u = (float)(hh >> 8) * 5.9604644775390625e-08f;
            const unsigned m32 = (unsigned)__ballot(cdfn > u);   // wave32: low 32 bits
            const int j = m32 ? (__ffs(m32) - 1) : 31;           // inverse-CDF pick
            const float pj = __shfl(nprob, j, 32);
            const int   nj = __shfl(nbr, j, 32);
            const float w  = 1.0f / (pj + EPSF);
            // wave-wide gather of one H row: 32 lanes x float4 = 512B coalesced
            const float4 hv = *((const float4*)(H + (size_t)nj * FIN + lane * 4));
            acc.x = fmaf(w, hv.x, acc.x);
            acc.y = fmaf(w, hv.y, acc.y);
            acc.z = fmaf(w, hv.z, acc.z);
            acc.w = fmaf(w, hv.w, acc.w);
        }
        const float inv_s = 1.0f / (float)SS;
        acc.x *= inv_s; acc.y *= inv_s; acc.z *= inv_s; acc.w *= inv_s;
        *((float4*)(agg + m * ASTR + lane * 4)) = acc;   // same-wave LDS, in order
    }

    // ---- Stage 2: D[16x128] = agg[16x128] * W[128x128] + b via WMMA f32 ----
    // A 16x4 frag: lanes 0-15 hold M=lane, K = kb,kb+1; lanes 16-31 K = kb+2,kb+3
    // B 4x16 frag: lanes 0-15 hold N=lane, rows kb,kb+1; lanes 16-31 rows kb+2,kb+3
    //   -> from transposed LDS this is one contiguous b64 per lane.
    // C/D 16x16  : VGPR v -> M = v (+8 for upper half-wave), N = lane&15
    const int mrow  = lane & 15;
    const int khalf = (lane >> 4) << 1;   // 0 or 2
    const int col   = lane & 15;

    v8f c[8];
#pragma unroll
    for (int n0 = 0; n0 < 8; ++n0) {
        const float bv = sB[n0 * 16 + col];   // bias broadcast across all M rows
#pragma unroll
        for (int v = 0; v < 8; ++v) c[n0][v] = bv;
    }

#pragma unroll 4
    for (int k0 = 0; k0 < FIN / 4; ++k0) {
        const int kb = 4 * k0 + khalf;
        const v2f a = *((const v2f*)(agg + mrow * ASTR + kb));     // ds_load_b64
#pragma unroll
        for (int n0 = 0; n0 < 8; ++n0) {
            const v2f bb = *((const v2f*)(sWt + (n0 * 16 + col) * TSTR + kb)); // ds_load_b64
            c[n0] = __builtin_amdgcn_wmma_f32_16x16x4_f32(
                /*neg_a=*/false, a, /*neg_b=*/false, bb,
                /*c_mod=*/(short)0, c[n0], /*reuse_a=*/false, /*reuse_b=*/false);
        }
    }

    // ---- Store ----
    const int rbase = tile * 16 + ((lane >> 4) << 3);
#pragma unroll
    for (int n0 = 0; n0 < 8; ++n0) {
#pragma unroll
        for (int v = 0; v < 8; ++v) {
            const int r = rbase + v;
            if (r < n) out[(size_t)r * FOUT + n0 * 16 + col] = c[n0][v];
        }
    }
}

// ---------------------------------------------------------------------------
extern "C" void kernel_launch(void* const* d_in, const int* in_sizes, int n_in,
                              void* d_out, int out_size, void* d_ws, size_t ws_size,
                              hipStream_t stream) {
    const float* H         = (const float*)d_in[0];   // [N, 128]
    const float* edge_vals = (const float*)d_in[1];   // [N, 32]
    const float* W         = (const float*)d_in[2];   // [128, 128]
    const float* b         = (const float*)d_in[3];   // [128]
    const int*   neighbors = (const int*)d_in[4];     // [N, 32]
    float* out = (float*)d_out;

    const int n = in_sizes[0] / FIN;

    // workspace: imp[n] | partials[nb1] | total[1]
    float* imp      = (float*)d_ws;
    const int nb1   = (n + 255) / 256;
    float* partials = imp + n;
    float* total    = partials + nb1;

    fgc_degree_kernel<<<nb1, 256, 0, stream>>>(edge_vals, imp, partials, n);
    fgc_reduce_kernel<<<1, 256, 0, stream>>>(partials, nb1, total);

    const int ntiles = (n + 15) / 16;
    const int blocks = (ntiles + WAVES_PER_BLOCK - 1) / WAVES_PER_BLOCK;
    (void)hipFuncSetAttribute((const void*)fgc_main_kernel,
                              hipFuncAttributeMaxDynamicSharedMemorySize,
                              (int)SMEM_BYTES);
    fgc_main_kernel<<<blocks, WAVES_PER_BLOCK * 32, SMEM_BYTES, stream>>>(
        H, neighbors, W, b, imp, total, out, n, ntiles);
}